// GraphConstruction_26800595927781
// MI455X (gfx1250) — compile-verified
//
#include <hip/hip_runtime.h>
#include <math.h>

// Problem constants (match reference): B=4, N=4096, D=64
#define BATCH 4
#define NN    4096
#define DD    64
#define LSTRIDE 4100   // padded LDS row stride in dwords: 16B-aligned, bank-conflict-free

typedef __attribute__((ext_vector_type(2))) float v2f;
typedef __attribute__((ext_vector_type(8))) float v8f;

// One workgroup = one (batch, 16-row) slab of the output.
// 8 wave32s; each wave owns 32 of the 256 column tiles.
// S slab (16 x 4096 fp32) lives in LDS (possible only with CDNA5's 320KB LDS),
// so every output byte is written exactly once: ~256MB total -> ~11us at 23.3TB/s.
__global__ __launch_bounds__(256) void graph_softmax_xxt(const float* __restrict__ X,
                                                         float* __restrict__ out) {
    __shared__ float sS[16 * LSTRIDE];   // 262,400 B
    __shared__ float sPM[256];           // partial maxima
    __shared__ float sPS[256];           // partial sums
    __shared__ float sInv[16];           // per-row 1/sum

    const int tid  = threadIdx.x;
    const int wave = tid >> 5;
    const int lane = tid & 31;
    const int lo   = lane & 15;   // row-within-tile this lane serves
    const int hi   = lane >> 4;   // selects K pair {0,1} vs {2,3} per ISA A/B layout

    const int b  = blockIdx.y;
    const int r0 = blockIdx.x << 4;

    const float* Xb = X + (size_t)b * NN * DD;

    // ---- Preload A fragments: lane L holds row (r0 + L%16); for k-step kk the
    // fragment is floats [4*kk + 2*hi, +1] of that row (ISA 16x4 f32 A layout).
    const float* arow = Xb + (size_t)(r0 + lo) * DD + 2 * hi;
    v2f afrag[16];
#pragma unroll
    for (int kk = 0; kk < 16; ++kk)
        afrag[kk] = *(const v2f*)(arow + 4 * kk);

    // ---- GEMM: S[r0:r0+16, :] = A · X^T via chained fp32 WMMA (K=4, 16 steps = D=64)
    for (int t = 0; t < 32; ++t) {
        const int j0 = (wave + (t << 3)) << 4;            // column tile base
        const float* brow = Xb + (size_t)(j0 + lo) * DD + 2 * hi;
        v8f c = {};
#pragma unroll
        for (int kk = 0; kk < 16; ++kk) {
            v2f bfrag = *(const v2f*)(brow + 4 * kk);
            c = __builtin_amdgcn_wmma_f32_16x16x4_f32(
                    /*neg_a=*/false, afrag[kk],
                    /*neg_b=*/false, bfrag,
                    /*c_mod=*/(short)0, c,
                    /*reuse_a=*/false, /*reuse_b=*/false);
        }
        // C/D layout: VGPR v, lanes 0-15 -> M=v, lanes 16-31 -> M=v+8; N = lane%16
#pragma unroll
        for (int v = 0; v < 8; ++v) {
            const int m = v + (hi << 3);
            sS[m * LSTRIDE + j0 + lo] = c[v];
        }
    }
    __syncthreads();

    // ---- Row softmax over the 16 x 4096 slab. 16 threads per row, 256 elems each.
    const int row = tid >> 4;
    const int seg = tid & 15;
    float* rowp = &sS[row * LSTRIDE + (seg << 8)];

    // pass 1: max
    float m = -INFINITY;
#pragma unroll 4
    for (int j = 0; j < 256; j += 4) {
        float4 v = *(const float4*)(rowp + j);
        m = fmaxf(m, fmaxf(fmaxf(v.x, v.y), fmaxf(v.z, v.w)));
    }
    sPM[tid] = m;
    __syncthreads();
    float M = -INFINITY;
#pragma unroll
    for (int k = 0; k < 16; ++k) M = fmaxf(M, sPM[(row << 4) + k]);

    // pass 2: exp in place + sum
    float s = 0.f;
#pragma unroll 4
    for (int j = 0; j < 256; j += 4) {
        float4 v = *(const float4*)(rowp + j);
        v.x = __expf(v.x - M);
        v.y = __expf(v.y - M);
        v.z = __expf(v.z - M);
        v.w = __expf(v.w - M);
        *(float4*)(rowp + j) = v;
        s += v.x + v.y + v.z + v.w;
    }
    sPS[tid] = s;
    __syncthreads();
    float S = 0.f;
#pragma unroll
    for (int k = 0; k < 16; ++k) S += sPS[(row << 4) + k];
    if (seg == 0) sInv[row] = 1.0f / S;
    __syncthreads();

    // ---- Single coalesced streaming write of the normalized slab (float4 stores)
    float* outBase = out + ((size_t)b * NN + r0) * (size_t)NN;
#pragma unroll 4
    for (int it = 0; it < 64; ++it) {
        const int fidx = tid + (it << 8);   // 0 .. 16383 float4s
        const int i  = fidx >> 10;          // row 0..15
        const int j4 = fidx & 1023;         // float4 index within row
        float4 v = *(const float4*)(&sS[i * LSTRIDE + (j4 << 2)]);
        const float inv = sInv[i];
        v.x *= inv; v.y *= inv; v.z *= inv; v.w *= inv;
        *(float4*)(outBase + (size_t)i * NN + (j4 << 2)) = v;
    }
}

extern "C" void kernel_launch(void* const* d_in, const int* in_sizes, int n_in,
                              void* d_out, int out_size, void* d_ws, size_t ws_size,
                              hipStream_t stream) {
    (void)in_sizes; (void)n_in; (void)d_ws; (void)ws_size; (void)out_size;
    const float* X = (const float*)d_in[0];
    float* out = (float*)d_out;
    dim3 grid(NN / 16, BATCH);   // 256 row-blocks x 4 batches = 1024 workgroups
    dim3 block(256);             // 8 wave32s
    graph_softmax_xxt<<<grid, block, 0, stream>>>(X, out);
}